// RetinaNetHead_48112223650601
// MI455X (gfx1250) — compile-verified
//
#include <hip/hip_runtime.h>
#include <hip/hip_bf16.h>
#include <stdint.h>

// ---------------- constants ----------------
#define NIMG     2
#define NBINS    8192      // 13-bit ordered-float histogram
#define CAP      4096      // per-(img,lvl) compaction capacity
#define PRE_N    1000
#define POST_N   100
#define KCAND    5000      // 5 levels * 1000
#define KPAD     8192
#define MASKW    80        // allocated u64 words per NMS row
#define WUSED    79        // ceil(5000/64)
#define NWAVE    8         // waves per 256-thread block
#define QOPS     4         // async b128 ops per pipeline stage (2 KB/stage/wave)
#define LOGIT_T  (-2.9444389791664403f)   // log(0.05/0.95): sigmoid(x)>0.05 <=> x>LOGIT_T
#define BOX_CLIP_C 4.135166556742356f

__device__ __constant__ int d_W[5]    = {128, 64, 32, 16, 8};          // h == w per level
__device__ __constant__ int d_AOFF[5] = {0, 147456, 184320, 193536, 195840};

// ordered-uint mapping of float (monotonic over all floats)
__device__ __forceinline__ unsigned fkey(float x) {
    unsigned u = __float_as_uint(x);
    return (u & 0x80000000u) ? ~u : (u | 0x80000000u);
}
__device__ __forceinline__ float keyf(unsigned k) {
    unsigned u = (k & 0x80000000u) ? (k ^ 0x80000000u) : ~k;
    return __uint_as_float(u);
}

// ---------------- K0: zero init ----------------
__global__ void zero32_kernel(unsigned* p, int n) {
    int i = blockIdx.x * blockDim.x + threadIdx.x;
    if (i < n) p[i] = 0u;
}

// ---------------- K1/K3: streaming scan over cls logits -----------
// mode 0: histogram of ordered keys (elements passing logit threshold)
// mode 1: compact (key,idx) pairs at/above cutoff bin
// Streams via a 2-deep ring of 4x GLOBAL_LOAD_ASYNC_TO_LDS_B128 groups
// per wave (4 KB in flight per wave, ASYNCcnt-tracked, in-order retire).
__global__ void scan_kernel(const float* __restrict__ cls, int lvl, int lg_hw,
                            int nchunks, int per_img,
                            unsigned* __restrict__ hist,
                            const unsigned* __restrict__ cutoffBin,
                            unsigned long long* __restrict__ cand,
                            unsigned* __restrict__ candCount, int mode)
{
    __shared__ float4 abuf[NWAVE][2][QOPS * 32];
    const int wid  = threadIdx.x >> 5;
    const int lane = threadIdx.x & 31;
    const int gw   = blockIdx.x * NWAVE + wid;
    const int nw   = gridDim.x * NWAVE;
    const int hw   = 1 << lg_hw;
    const float4* base = (const float4*)cls;

    // hoisted cutoffs (contents only meaningful in mode 1; harmless read otherwise)
    const unsigned cut0 = cutoffBin[lvl];
    const unsigned cut1 = cutoffBin[5 + lvl];

    auto issueGroup = [&](int chunk, int sel) {
#pragma unroll
        for (int q = 0; q < QOPS; ++q) {
            const float4* p = base + (size_t)chunk * (QOPS * 32) + q * 32 + lane;
            unsigned loff   = (unsigned)(size_t)&abuf[wid][sel][q * 32 + lane];
            uint64_t addr   = (uint64_t)(size_t)p;
            asm volatile("global_load_async_to_lds_b128 %0, %1, off"
                         :: "v"(loff), "v"(addr) : "memory");
        }
    };

    int sel = 0;
    if (gw < nchunks) issueGroup(gw, 0);
    for (int c = gw; c < nchunks; c += nw, sel ^= 1) {
        int cn = c + nw;
        if (cn < nchunks) {
            issueGroup(cn, sel ^ 1);
            // 8 outstanding, in-order completion: cnt<=4 => current group done
            asm volatile("s_wait_asynccnt 4" ::: "memory");
        } else {
            asm volatile("s_wait_asynccnt 0" ::: "memory");
        }
#pragma unroll
        for (int q = 0; q < QOPS; ++q) {
            float4 v4 = abuf[wid][sel][q * 32 + lane];
            int fbase = (c * (QOPS * 32) + q * 32 + lane) * 4;
            float vv[4] = {v4.x, v4.y, v4.z, v4.w};
#pragma unroll
            for (int k = 0; k < 4; ++k) {
                float v = vv[k];
                if (v > LOGIT_T) {
                    unsigned key = fkey(v);
                    int f    = fbase + k;
                    int img  = (f >= per_img) ? 1 : 0;          // N == 2
                    int pair = img * 5 + lvl;
                    if (mode == 0) {
                        atomicAdd(&hist[pair * NBINS + (key >> 19)], 1u);
                    } else {
                        unsigned cut = img ? cut1 : cut0;
                        if ((key >> 19) >= cut) {
                            unsigned slot = atomicAdd(&candCount[pair], 1u);
                            if (slot < CAP) {
                                int rem = f - img * per_img;    // layout [720][h][w]
                                int ch  = rem >> lg_hw;
                                int pix = rem & (hw - 1);
                                // flattened candidate idx = pix*A*C + ch (matches reference)
                                cand[(size_t)pair * CAP + slot] =
                                    ((unsigned long long)key << 32) |
                                    (unsigned)(pix * 720 + ch);
                            }
                        }
                    }
                }
            }
        }
    }
}

// ---------------- K2: find cutoff bin (rank PRE_N) ----------------
__global__ void cutoff_kernel(const unsigned* __restrict__ hist,
                              unsigned* __restrict__ cutoffBin)
{
    int pair = blockIdx.x;
    if (threadIdx.x == 0) {
        const unsigned* h = hist + (size_t)pair * NBINS;
        unsigned acc = 0, cut = 0;
        for (int b = NBINS - 1; b >= 0; --b) {
            acc += h[b];
            if (acc >= (unsigned)PRE_N) { cut = (unsigned)b; break; }
        }
        cutoffBin[pair] = cut;   // stays 0 if fewer than PRE_N pass (take all)
    }
}

// ---------------- shared bitonic (ascending, n power of two) ------
__device__ __forceinline__ void bitonic_asc(unsigned long long* s, int n) {
    for (unsigned k = 2; k <= (unsigned)n; k <<= 1) {
        for (unsigned j = k >> 1; j > 0; j >>= 1) {
            for (unsigned i = threadIdx.x; i < (unsigned)n; i += blockDim.x) {
                unsigned ixj = i ^ j;
                if (ixj > i) {
                    unsigned long long a = s[i], b = s[ixj];
                    if (((i & k) == 0) ? (a > b) : (a < b)) { s[i] = b; s[ixj] = a; }
                }
            }
            __syncthreads();
        }
    }
}

// ---------------- K4: per-(img,lvl) sort -> top-1000 --------------
__global__ void select_pair_kernel(const unsigned long long* __restrict__ cand,
                                   const unsigned* __restrict__ candCount,
                                   unsigned long long* __restrict__ perKey)
{
    __shared__ unsigned long long s[CAP];
    int pair = blockIdx.x, img = pair / 5, lvl = pair % 5;
    unsigned cnt = candCount[pair];
    if (cnt > CAP) cnt = CAP;
    for (int i = threadIdx.x; i < CAP; i += blockDim.x)
        s[i] = (i < (int)cnt) ? cand[(size_t)pair * CAP + i] : 0ull;
    __syncthreads();
    bitonic_asc(s, CAP);
    for (int r = threadIdx.x; r < PRE_N; r += blockDim.x) {
        unsigned long long e = s[CAP - 1 - r];
        unsigned long long o = 0ull;
        if (e) {
            unsigned key = (unsigned)(e >> 32);
            unsigned idx = (unsigned)e;                 // < 2^24
            o = ((unsigned long long)key << 32) | (idx | ((unsigned)lvl << 27));
        }
        perKey[(size_t)img * KCAND + lvl * PRE_N + r] = o;
    }
}

// ---------------- K5: per-image sort(5000) + lazy decode ----------
__global__ void image_sort_decode(const unsigned long long* __restrict__ perKey,
                                  const float* __restrict__ b0, const float* __restrict__ b1,
                                  const float* __restrict__ b2, const float* __restrict__ b3,
                                  const float* __restrict__ b4,
                                  const float* __restrict__ anchors,
                                  const int* __restrict__ image_sizes,
                                  float* __restrict__ sBox, float* __restrict__ sScore,
                                  int* __restrict__ sCls)
{
    __shared__ unsigned long long s[KPAD];   // 64 KB
    int img = blockIdx.x;
    for (int i = threadIdx.x; i < KPAD; i += blockDim.x)
        s[i] = (i < KCAND) ? perKey[(size_t)img * KCAND + i] : 0ull;
    __syncthreads();
    bitonic_asc(s, KPAD);

    const float* bp[5] = {b0, b1, b2, b3, b4};
    float Himg = (float)image_sizes[img * 2 + 0];
    float Wimg = (float)image_sizes[img * 2 + 1];

    for (int r = threadIdx.x; r < KCAND; r += blockDim.x) {
        unsigned long long e = s[KPAD - 1 - r];
        float4 ob = {0.f, 0.f, 0.f, 0.f};
        float  sc = -1.f;
        int    lab = 0;
        if (e) {
            unsigned key = (unsigned)(e >> 32);
            unsigned low = (unsigned)e;
            int lvl = (low >> 27) & 7;
            int idx = (int)(low & ((1u << 27) - 1));
            float logit = keyf(key);
            sc = 1.f / (1.f + expf(-logit));
            int pix = idx / 720, ch = idx - pix * 720;
            int a = ch / 80, c = ch - a * 80;
            int w = d_W[lvl];                      // h == w
            int y = pix / w, x = pix - y * w;
            int g = d_AOFF[lvl] + pix * 9 + a;
            const float* bpl = bp[lvl];
            size_t pb = ((size_t)img * 36 + a * 4) * w * w + (size_t)y * w + x;
            size_t hwl = (size_t)w * w;
            float dx = bpl[pb];
            float dy = bpl[pb + hwl];
            float dw = bpl[pb + 2 * hwl];
            float dh = bpl[pb + 3 * hwl];
            float ax0 = anchors[g * 4 + 0], ay0 = anchors[g * 4 + 1];
            float ax1 = anchors[g * 4 + 2], ay1 = anchors[g * 4 + 3];
            float aw = ax1 - ax0, ah = ay1 - ay0;
            float acx = ax0 + 0.5f * aw, acy = ay0 + 0.5f * ah;
            dw = fminf(dw, BOX_CLIP_C);
            dh = fminf(dh, BOX_CLIP_C);
            float pcx = dx * aw + acx, pcy = dy * ah + acy;
            float pw = expf(dw) * aw, ph = expf(dh) * ah;
            ob.x = fminf(fmaxf(pcx - 0.5f * pw, 0.f), Wimg);
            ob.y = fminf(fmaxf(pcy - 0.5f * ph, 0.f), Himg);
            ob.z = fminf(fmaxf(pcx + 0.5f * pw, 0.f), Wimg);
            ob.w = fminf(fmaxf(pcy + 0.5f * ph, 0.f), Himg);
            lab = c;
        }
        size_t o = (size_t)img * KCAND + r;
        *(float4*)&sBox[o * 4] = ob;
        sScore[o] = sc;
        sCls[o]   = lab;
    }
}

// ---------------- K6: NMS suppression bitmask ---------------------
// class-aware: CLASS_OFFSET > max coord => cross-class IoU is exactly 0,
// so only same-class pairs are tested.
__global__ void nms_mask_kernel(const float* __restrict__ sBox,
                                const float* __restrict__ sScore,
                                const int* __restrict__ sCls,
                                unsigned long long* __restrict__ mask)
{
    __shared__ float4 cB[64];
    __shared__ float  cS[64];
    __shared__ int    cC[64];
    int img = blockIdx.z, cb = blockIdx.x, rb = blockIdx.y, t = threadIdx.x;
    int j = cb * 64 + t;
    if (j < KCAND) {
        size_t o = (size_t)img * KCAND + j;
        cB[t] = *(const float4*)&sBox[o * 4];
        cS[t] = sScore[o];
        cC[t] = sCls[o];
    } else {
        cS[t] = -1.f; cC[t] = -1; cB[t] = make_float4(0.f, 0.f, 0.f, 0.f);
    }
    __syncthreads();

    int i = rb * 64 + t;
    if (i >= KCAND) return;
    size_t oi = (size_t)img * KCAND + i;
    float4 bi = *(const float4*)&sBox[oi * 4];
    float  si = sScore[oi];
    int    ci = sCls[oi];
    unsigned long long bits = 0ull;
    if (si > 0.f) {
        float areai = (bi.z - bi.x) * (bi.w - bi.y);
        for (int jj = 0; jj < 64; ++jj) {
            int j2 = cb * 64 + jj;
            if (j2 >= KCAND || j2 == i) continue;
            if (cS[jj] <= 0.f || cC[jj] != ci) continue;
            float4 bj = cB[jj];
            float ix1 = fmaxf(bi.x, bj.x), iy1 = fmaxf(bi.y, bj.y);
            float ix2 = fminf(bi.z, bj.z), iy2 = fminf(bi.w, bj.w);
            float inter = fmaxf(ix2 - ix1, 0.f) * fmaxf(iy2 - iy1, 0.f);
            float areaj = (bj.z - bj.x) * (bj.w - bj.y);
            float iou = inter / (areai + areaj - inter);   // 0/0 -> NaN -> not > 0.5 (matches ref)
            if (iou > 0.5f) bits |= 1ull << jj;
        }
    }
    mask[((size_t)img * KCAND + i) * MASKW + cb] = bits;
}

// ---------------- K7: sequential NMS reduce + emit top-100 --------
__global__ void nms_reduce_kernel(const float* __restrict__ sBox,
                                  const float* __restrict__ sScore,
                                  const int* __restrict__ sCls,
                                  const unsigned long long* __restrict__ mask,
                                  float* __restrict__ out)
{
    __shared__ unsigned long long rem[WUSED];
    __shared__ int kept;
    __shared__ int take;
    int img = blockIdx.x, tid = threadIdx.x;
    if (tid < WUSED) rem[tid] = 0ull;
    if (tid == 0) kept = 0;
    __syncthreads();
    for (int i = 0; i < KCAND; ++i) {
        if (tid == 0) {
            bool v = (sScore[(size_t)img * KCAND + i] > 0.f) &&
                     !((rem[i >> 6] >> (i & 63)) & 1ull);
            take = v ? 1 : 0;
        }
        __syncthreads();
        if (take) {
            const unsigned long long* mr = &mask[((size_t)img * KCAND + i) * MASKW];
            for (int t = tid; t < WUSED; t += blockDim.x) rem[t] |= mr[t];
            if (tid == 0) {
                if (kept < POST_N) {
                    int r = kept;
                    size_t o = (size_t)img * KCAND + i;
                    out[((size_t)img * POST_N + r) * 4 + 0] = sBox[o * 4 + 0];
                    out[((size_t)img * POST_N + r) * 4 + 1] = sBox[o * 4 + 1];
                    out[((size_t)img * POST_N + r) * 4 + 2] = sBox[o * 4 + 2];
                    out[((size_t)img * POST_N + r) * 4 + 3] = sBox[o * 4 + 3];
                    out[NIMG * POST_N * 4 + img * POST_N + r]                 = sScore[o];
                    out[NIMG * POST_N * 4 + NIMG * POST_N + img * POST_N + r] = (float)(sCls[o] + 1);
                }
                kept++;
            }
        }
        __syncthreads();
    }
}

// ---------------- host launch -------------------------------------
extern "C" void kernel_launch(void* const* d_in, const int* in_sizes, int n_in,
                              void* d_out, int out_size, void* d_ws, size_t ws_size,
                              hipStream_t stream) {
    (void)in_sizes; (void)n_in; (void)out_size; (void)ws_size;
    const float* cls[5]  = {(const float*)d_in[0], (const float*)d_in[2], (const float*)d_in[4],
                            (const float*)d_in[6], (const float*)d_in[8]};
    const float* boxp[5] = {(const float*)d_in[1], (const float*)d_in[3], (const float*)d_in[5],
                            (const float*)d_in[7], (const float*)d_in[9]};
    const float* anchors = (const float*)d_in[10];
    const int*   isz     = (const int*)d_in[11];

    uint8_t* p = (uint8_t*)d_ws;
    auto take = [&](size_t bytes) -> void* {
        void* r = (void*)p;
        p += (bytes + 255) & ~(size_t)255;
        return r;
    };
    unsigned*            hist    = (unsigned*)take((size_t)10 * NBINS * 4);
    unsigned*            cutoff  = (unsigned*)take(64);
    unsigned*            ccount  = (unsigned*)take(64);
    unsigned long long*  cand    = (unsigned long long*)take((size_t)10 * CAP * 8);
    unsigned long long*  perKey  = (unsigned long long*)take((size_t)NIMG * KCAND * 8);
    float*               sBox    = (float*)take((size_t)NIMG * KCAND * 4 * 4);
    float*               sScore  = (float*)take((size_t)NIMG * KCAND * 4);
    int*                 sCls    = (int*)take((size_t)NIMG * KCAND * 4);
    unsigned long long*  mask    = (unsigned long long*)take((size_t)NIMG * KCAND * MASKW * 8);

    // zero state used this call (workspace is not re-poisoned between replays)
    zero32_kernel<<<(10 * NBINS + 255) / 256, 256, 0, stream>>>(hist, 10 * NBINS);
    zero32_kernel<<<1, 32, 0, stream>>>(ccount, 16);
    zero32_kernel<<<(1200 + 255) / 256, 256, 0, stream>>>((unsigned*)d_out, 1200);

    static const int LGHW[5] = {14, 12, 10, 8, 6};   // log2(h*w)
    // pass 1: histogram
    for (int l = 0; l < 5; ++l) {
        int hw = 1 << LGHW[l];
        int per_img = 720 * hw;
        int nchunks = (2 * per_img) / (4 * 32 * QOPS);   // exact for all levels
        int blocks = (nchunks + NWAVE - 1) / NWAVE;
        if (blocks > 1024) blocks = 1024;
        scan_kernel<<<blocks, 256, 0, stream>>>(cls[l], l, LGHW[l], nchunks, per_img,
                                                hist, cutoff, cand, ccount, 0);
    }
    cutoff_kernel<<<10, 32, 0, stream>>>(hist, cutoff);
    // pass 2: compact
    for (int l = 0; l < 5; ++l) {
        int hw = 1 << LGHW[l];
        int per_img = 720 * hw;
        int nchunks = (2 * per_img) / (4 * 32 * QOPS);
        int blocks = (nchunks + NWAVE - 1) / NWAVE;
        if (blocks > 1024) blocks = 1024;
        scan_kernel<<<blocks, 256, 0, stream>>>(cls[l], l, LGHW[l], nchunks, per_img,
                                                hist, cutoff, cand, ccount, 1);
    }
    select_pair_kernel<<<10, 512, 0, stream>>>(cand, ccount, perKey);
    image_sort_decode<<<NIMG, 512, 0, stream>>>(perKey, boxp[0], boxp[1], boxp[2], boxp[3],
                                                boxp[4], anchors, isz, sBox, sScore, sCls);
    dim3 g((KCAND + 63) / 64, (KCAND + 63) / 64, NIMG);
    nms_mask_kernel<<<g, 64, 0, stream>>>(sBox, sScore, sCls, mask);
    nms_reduce_kernel<<<NIMG, 128, 0, stream>>>(sBox, sScore, sCls, mask, (float*)d_out);
}